// Encoder_83176336655079
// MI455X (gfx1250) — compile-verified
//
#include <hip/hip_runtime.h>
#include <hip/hip_bf16.h>
#include <math.h>

// Problem dims (fixed by reference)
#define Vv   32000
#define Ee   512
#define Hh   1024
#define Bb   128
#define Tt   256
#define G4H  4096            // 4*H
#define KTOT 1536            // E + H

// Tiling
#define ASTRIDE       1540   // padded LDS row stride (floats): 1540 % 64 = 4 -> conflict free
#define NBLK_PER_DIR  128    // 8 m-tiles * 16 col-groups
#define THREADS       256

#define LDS_A_FLOATS  (16 * ASTRIDE)   // 24640
#define LDS_Z_FLOATS  (16 * 256)       // 4096
#define LDS_TOTAL_FLOATS (LDS_A_FLOATS + LDS_Z_FLOATS + 16)
#define LDS_BYTES     (LDS_TOTAL_FLOATS * 4)   // 115,008 B

typedef __attribute__((ext_vector_type(2))) float v2f;
typedef __attribute__((ext_vector_type(8))) float v8f;

__device__ __forceinline__ float sigmoidf_(float x) {
    return 1.0f / (1.0f + __expf(-x));
}

// Async memory->LDS copy of 16B: VDST operand = LDS byte offset (dynamic LDS
// starts at 0: kernel has no static __shared__), VADDR = 64-bit global addr.
// Tracked by ASYNCcnt; no VGPR data round-trip.
__device__ __forceinline__ void async_ld_b128(int lds_off, const void* gptr) {
    unsigned long long ga = (unsigned long long)(uintptr_t)gptr;
    asm volatile("global_load_async_to_lds_b128 %0, %1, off"
                 :: "v"(lds_off), "v"(ga) : "memory");
}
__device__ __forceinline__ void wait_asynccnt0() {
    asm volatile("s_wait_asynccnt 0x0" ::: "memory");
}

// ---------------------------------------------------------------------------
// init: zero h ping-pong buffers and barrier counters (re-done every launch
// -> deterministic for graph replay).
// ---------------------------------------------------------------------------
__global__ void bilstm_init(float* __restrict__ hbuf_f,
                            float* __restrict__ hbuf_b,
                            unsigned* __restrict__ counters) {
    size_t i = (size_t)blockIdx.x * blockDim.x + threadIdx.x;
    const size_t n = 2ull * Bb * Hh;   // two ping-pong planes each
    if (i < n) { hbuf_f[i] = 0.0f; hbuf_b[i] = 0.0f; }
    if (i < 8) counters[i] = 0u;
}

// ---------------------------------------------------------------------------
// Persistent bidirectional-LSTM kernel.
//   grid = 256 blocks: blockIdx[7] = direction, then 8 m-tiles x 16 col-groups
//   block = 256 threads = 8 waves (wave32)
// Each block owns batch rows [m0,m0+16) and h-columns [hc0,hc0+64).
// Wave w (= gate*2 + half) computes the 16x32 z slice for
//   weight cols gate*1024 + hc0 + half*32 + [0,32)  via f32 WMMA 16x16x4.
// ---------------------------------------------------------------------------
__global__ void __launch_bounds__(THREADS, 1) bilstm_persistent(
    const int*   __restrict__ x,
    const float* __restrict__ emb,
    const float* __restrict__ w_f, const float* __restrict__ u_f, const float* __restrict__ b_f,
    const float* __restrict__ w_b, const float* __restrict__ u_b, const float* __restrict__ b_b,
    float* __restrict__ dout,      // ys_f written here directly
    float* __restrict__ ysb,       // ys_b workspace
    float* __restrict__ hbuf_f,    // 2 * B*H ping-pong
    float* __restrict__ hbuf_b,    // 2 * B*H ping-pong
    float* __restrict__ hfin_f, float* __restrict__ cfin_f,
    float* __restrict__ hfin_b, float* __restrict__ cfin_b,
    unsigned* __restrict__ counters)
{
    extern __shared__ float smem[];
    float* ldsA  = smem;                        // 16 x ASTRIDE : [e_t | h] rows
    float* zbuf  = smem + LDS_A_FLOATS;         // 16 x 256 gate pre-activations
    float* maskf = zbuf + LDS_Z_FLOATS;         // 16 row masks

    const int dir   = blockIdx.x >> 7;
    const int bid   = blockIdx.x & 127;
    const int m0    = (bid >> 4) * 16;          // batch row base
    const int hc0   = (bid & 15) * 64;          // h-column base

    const float* W   = dir ? w_b : w_f;
    const float* U   = dir ? u_b : u_f;
    const float* Bv  = dir ? b_b : b_f;
    float* hbuf      = dir ? hbuf_b : hbuf_f;
    float* ybase     = dir ? ysb    : dout;
    float* hfin      = dir ? hfin_b : hfin_f;
    float* cfin      = dir ? cfin_b : cfin_f;
    unsigned* ctr    = counters + dir;

    const int tid    = threadIdx.x;
    const int lane   = tid & 31;
    const int wv     = tid >> 5;
    const int laneHi = lane >> 4;               // 0|1 -> K sub-offset 0|2
    const int ln     = lane & 15;

    // wave role: which 32 z-columns (two 16-wide WMMA tiles)
    const int gate     = wv >> 1;               // 0=i 1=f 2=g 3=o
    const int half     = wv & 1;
    const int zc0      = gate * 64 + half * 32;              // local zbuf col
    const int ng0      = gate * 1024 + hc0 + half * 32 + ln; // global weight col
    const float bias0  = Bv[ng0];
    const float bias1  = Bv[ng0 + 16];
    const float* wp0 = W + ng0;  const float* wp1 = W + ng0 + 16;
    const float* up0 = U + ng0;  const float* up1 = U + ng0 + 16;

    // combine-phase ownership: this thread owns (row=crow, cols ccol..ccol+3)
    const int crow = tid >> 4;                  // 0..15
    const int ccol = (tid & 15) * 4;            // 0..60
    float creg[4] = {0.f, 0.f, 0.f, 0.f};       // cell state (register-resident)
    float hreg[4] = {0.f, 0.f, 0.f, 0.f};       // hidden state

    for (int s = 0; s < Tt; ++s) {
        const int t = dir ? (Tt - 1 - s) : s;
        const float* hsrc = hbuf + (size_t)(s & 1) * (Bb * Hh);
        float*       hdst = hbuf + (size_t)((s + 1) & 1) * (Bb * Hh);

        // keep weight streams warm in L2 (global_prefetch_b8)
        __builtin_prefetch(wp0, 0, 1);
        __builtin_prefetch(up0, 0, 1);

        // ---- stage A (async -> LDS): embedding rows e[b,t,:] (16 x 512) ----
        #pragma unroll
        for (int i = 0; i < 8; ++i) {
            int idx = tid + i * THREADS;        // 0..2047 float4 slots
            int row = idx >> 7;
            int kq  = idx & 127;
            int tok = x[(m0 + row) * Tt + t];
            async_ld_b128((row * ASTRIDE + kq * 4) * 4,
                          emb + (size_t)tok * Ee + kq * 4);
        }
        // ---- stage A (async -> LDS): hidden rows h[b,:] (16 x 1024) ----
        #pragma unroll
        for (int i = 0; i < 16; ++i) {
            int idx = tid + i * THREADS;        // 0..4095 float4 slots
            int row = idx >> 8;
            int kq  = idx & 255;
            async_ld_b128((row * ASTRIDE + Ee + kq * 4) * 4,
                          hsrc + (size_t)(m0 + row) * Hh + kq * 4);
        }
        if (tid < 16) {
            int tok = x[(m0 + tid) * Tt + t];
            maskf[tid] = (tok != 0) ? 1.0f : 0.0f;
        }
        wait_asynccnt0();        // this wave's async LDS writes have landed
        __syncthreads();         // everyone's have

        // ---- GEMM: z = [e|h] @ [W;U] + b   (per wave: 16x32, K=1536) ----
        v8f acc0, acc1;
        #pragma unroll
        for (int i = 0; i < 8; ++i) { acc0[i] = bias0; acc1[i] = bias1; }

        const float* aBase = ldsA + ln * ASTRIDE;

        #pragma unroll 4
        for (int k0 = 0; k0 < Ee; k0 += 4) {
            int kk = k0 + laneHi * 2;
            v2f a = *(const v2f*)(aBase + kk);          // A[row=ln, kk..kk+1]
            size_t off = (size_t)kk * G4H;
            v2f b0; b0.x = wp0[off]; b0.y = wp0[off + G4H];
            v2f b1; b1.x = wp1[off]; b1.y = wp1[off + G4H];
            acc0 = __builtin_amdgcn_wmma_f32_16x16x4_f32(false, a, false, b0,
                                                         (short)0, acc0, false, false);
            acc1 = __builtin_amdgcn_wmma_f32_16x16x4_f32(false, a, false, b1,
                                                         (short)0, acc1, false, false);
        }
        #pragma unroll 4
        for (int k0 = 0; k0 < Hh; k0 += 4) {
            int kk = k0 + laneHi * 2;
            v2f a = *(const v2f*)(aBase + Ee + kk);
            size_t off = (size_t)kk * G4H;
            v2f b0; b0.x = up0[off]; b0.y = up0[off + G4H];
            v2f b1; b1.x = up1[off]; b1.y = up1[off + G4H];
            acc0 = __builtin_amdgcn_wmma_f32_16x16x4_f32(false, a, false, b0,
                                                         (short)0, acc0, false, false);
            acc1 = __builtin_amdgcn_wmma_f32_16x16x4_f32(false, a, false, b1,
                                                         (short)0, acc1, false, false);
        }

        // ---- spill z tiles to LDS (C layout: vgpr r -> row r / r+8) ----
        const int zrow0 = laneHi * 8;
        #pragma unroll
        for (int r = 0; r < 8; ++r) {
            zbuf[(zrow0 + r) * 256 + zc0 + ln]      = acc0[r];
            zbuf[(zrow0 + r) * 256 + zc0 + 16 + ln] = acc1[r];
        }
        __syncthreads();

        // ---- gate fusion + state update (4 h-cols per thread) ----
        {
            const float m  = maskf[crow];
            const float* zr = zbuf + crow * 256;
            float4 zi = *(const float4*)(zr +   0 + ccol);
            float4 zf = *(const float4*)(zr +  64 + ccol);
            float4 zg = *(const float4*)(zr + 128 + ccol);
            float4 zo = *(const float4*)(zr + 192 + ccol);
            float zia[4] = {zi.x, zi.y, zi.z, zi.w};
            float zfa[4] = {zf.x, zf.y, zf.z, zf.w};
            float zga[4] = {zg.x, zg.y, zg.z, zg.w};
            float zoa[4] = {zo.x, zo.y, zo.z, zo.w};
            #pragma unroll
            for (int j = 0; j < 4; ++j) {
                float cn = sigmoidf_(zfa[j]) * creg[j] +
                           sigmoidf_(zia[j]) * tanhf(zga[j]);
                float hn = sigmoidf_(zoa[j]) * tanhf(cn);
                creg[j] = m * cn + (1.0f - m) * creg[j];
                hreg[j] = m * hn + (1.0f - m) * hreg[j];
            }
            float4 h4 = make_float4(hreg[0], hreg[1], hreg[2], hreg[3]);
            const int gb = m0 + crow;
            const int gc = hc0 + ccol;
            *(float4*)(hdst + (size_t)gb * Hh + gc) = h4;                 // publish h
            *(float4*)(ybase + ((size_t)gb * Tt + t) * Hh + gc) = h4;     // ys output
        }

        // ---- grid-wide barrier (per direction); ping-pong h removes WAR ----
        __threadfence();
        __syncthreads();
        if (tid == 0) {
            __hip_atomic_fetch_add(ctr, 1u, __ATOMIC_ACQ_REL, __HIP_MEMORY_SCOPE_AGENT);
            const unsigned target = (unsigned)NBLK_PER_DIR * (unsigned)(s + 1);
            while (__hip_atomic_load(ctr, __ATOMIC_ACQUIRE, __HIP_MEMORY_SCOPE_AGENT) < target) {
                __builtin_amdgcn_s_sleep(2);
            }
        }
        __syncthreads();
    }

    // ---- final states ----
    {
        const int gb = m0 + crow;
        const int gc = hc0 + ccol;
        *(float4*)(hfin + (size_t)gb * Hh + gc) =
            make_float4(hreg[0], hreg[1], hreg[2], hreg[3]);
        *(float4*)(cfin + (size_t)gb * Hh + gc) =
            make_float4(creg[0], creg[1], creg[2], creg[3]);
    }
}

// ---------------------------------------------------------------------------
// combine: out += ys_b ; out tail = (h_f+h_b, c_f+c_b)
// ---------------------------------------------------------------------------
__global__ void bilstm_combine(float* __restrict__ dout,
                               const float* __restrict__ ysb,
                               const float* __restrict__ hfin_f,
                               const float* __restrict__ cfin_f,
                               const float* __restrict__ hfin_b,
                               const float* __restrict__ cfin_b) {
    size_t i = (size_t)blockIdx.x * blockDim.x + threadIdx.x;
    const size_t N4  = (size_t)Bb * Tt * Hh / 4;   // float4 count of ys
    const size_t BH4 = (size_t)Bb * Hh / 4;
    if (i < N4) {
        float4 a = ((const float4*)dout)[i];
        float4 b = ((const float4*)ysb)[i];
        a.x += b.x; a.y += b.y; a.z += b.z; a.w += b.w;
        ((float4*)dout)[i] = a;
    }
    if (i < BH4) {
        float4 hf = ((const float4*)hfin_f)[i], hb = ((const float4*)hfin_b)[i];
        float4 cf = ((const float4*)cfin_f)[i], cb = ((const float4*)cfin_b)[i];
        ((float4*)dout)[N4 + i] =
            make_float4(hf.x + hb.x, hf.y + hb.y, hf.z + hb.z, hf.w + hb.w);
        ((float4*)dout)[N4 + BH4 + i] =
            make_float4(cf.x + cb.x, cf.y + cb.y, cf.z + cb.z, cf.w + cb.w);
    }
}

// ---------------------------------------------------------------------------
extern "C" void kernel_launch(void* const* d_in, const int* in_sizes, int n_in,
                              void* d_out, int out_size, void* d_ws, size_t ws_size,
                              hipStream_t stream) {
    const int*   x    = (const int*)  d_in[0];
    const float* emb  = (const float*)d_in[1];
    const float* w_f  = (const float*)d_in[2];
    const float* u_f  = (const float*)d_in[3];
    const float* b_f  = (const float*)d_in[4];
    const float* w_b  = (const float*)d_in[5];
    const float* u_b  = (const float*)d_in[6];
    const float* b_b  = (const float*)d_in[7];
    float* dout = (float*)d_out;

    const size_t BTH = (size_t)Bb * Tt * Hh;
    const size_t BH  = (size_t)Bb * Hh;

    float* ws     = (float*)d_ws;
    float* ysb    = ws;                      // B*T*H
    float* hbuf_f = ysb + BTH;               // 2*B*H (ping-pong)
    float* hbuf_b = hbuf_f + 2 * BH;         // 2*B*H
    float* hfin_f = hbuf_b + 2 * BH;
    float* cfin_f = hfin_f + BH;
    float* hfin_b = cfin_f + BH;
    float* cfin_b = hfin_b + BH;
    unsigned* counters = (unsigned*)(cfin_b + BH);

    hipLaunchKernelGGL(bilstm_init, dim3(1024), dim3(THREADS), 0, stream,
                       hbuf_f, hbuf_b, counters);

    hipLaunchKernelGGL(bilstm_persistent, dim3(2 * NBLK_PER_DIR), dim3(THREADS),
                       LDS_BYTES, stream,
                       x, emb, w_f, u_f, b_f, w_b, u_b, b_b,
                       dout, ysb, hbuf_f, hbuf_b,
                       hfin_f, cfin_f, hfin_b, cfin_b, counters);

    const size_t N4 = BTH / 4;
    hipLaunchKernelGGL(bilstm_combine, dim3((unsigned)((N4 + THREADS - 1) / THREADS)),
                       dim3(THREADS), 0, stream,
                       dout, ysb, hfin_f, cfin_f, hfin_b, cfin_b);
}